// GaussianMixture_50732153701067
// MI455X (gfx1250) — compile-verified
//
#include <hip/hip_runtime.h>
#include <math.h>

typedef __attribute__((ext_vector_type(2))) float v2f;
typedef __attribute__((ext_vector_type(8))) float v8f;

#define N_PTS   1000000
#define M_COMP  64
#define ITERS   4
#define INV_LN2 1.4426950408889634f
#define LN2f    0.6931471805599453f

// ---------------------------------------------------------------------------
// Prep: build per-component rank-8 vectors b_j (scaled by 1/ln2) into d_ws.
//   b_j = (1/ln2) * [-g00,-g01,-g10,-g11, 2*(G mu)_0, 2*(G mu)_1,
//                    wlog_j - mu^T G mu, 0]
// where wlog_j = log_softmax(w)_j + 0.5*log(det G_j),  G = A A^T / 2.
// ---------------------------------------------------------------------------
__global__ __launch_bounds__(64)
void gmm_prep(const float* __restrict__ mu,
              const float* __restrict__ A,
              const float* __restrict__ w,
              float* __restrict__ Bt) {
    __shared__ float red[M_COMP];
    const int j = threadIdx.x;
    const float wj = w[j];
    red[j] = wj;
    __syncthreads();
    float m = red[0];
    #pragma unroll
    for (int i = 1; i < M_COMP; ++i) m = fmaxf(m, red[i]);
    float s = 0.0f;
    #pragma unroll
    for (int i = 0; i < M_COMP; ++i) s += expf(red[i] - m);
    const float lse = m + logf(s);

    const float a00 = A[j*4+0], a01 = A[j*4+1], a10 = A[j*4+2], a11 = A[j*4+3];
    const float g00 = 0.5f*(a00*a00 + a01*a01);
    const float g01 = 0.5f*(a00*a10 + a01*a11);   // == g10
    const float g11 = 0.5f*(a10*a10 + a11*a11);
    const float det = g00*g11 - g01*g01;

    const float m0 = mu[j*2+0], m1 = mu[j*2+1];
    const float Gm0 = g00*m0 + g01*m1;
    const float Gm1 = g01*m0 + g11*m1;
    const float cst = m0*Gm0 + m1*Gm1;

    const float wlog = (wj - lse) + 0.5f*logf(det);

    Bt[j*8+0] = -g00 * INV_LN2;
    Bt[j*8+1] = -g01 * INV_LN2;
    Bt[j*8+2] = -g01 * INV_LN2;
    Bt[j*8+3] = -g11 * INV_LN2;
    Bt[j*8+4] =  2.0f*Gm0 * INV_LN2;
    Bt[j*8+5] =  2.0f*Gm1 * INV_LN2;
    Bt[j*8+6] = (wlog - cst) * INV_LN2;
    Bt[j*8+7] = 0.0f;
}

// ---------------------------------------------------------------------------
// Main: D = A x B with A = components (M), B = points (N).
// C/D layout: lane = point (N=l16), VGPR r = component (M = 16t + r + 8*half)
//  -> per-point logsumexp over 64 components is 32 in-lane values + one
//     shfl_xor(16) cross-half merge for max, and again for the sum.
// Each wave owns ITERS consecutive 16-point tiles; the component operands
// (A side of the WMMA) are loop-invariant and hoisted.
// ---------------------------------------------------------------------------
__global__ __launch_bounds__(256)
void gmm_main(const float* __restrict__ sample,
              const float* __restrict__ Bt,
              float* __restrict__ out) {
    const int lane  = threadIdx.x & 31;
    const int wave  = threadIdx.x >> 5;
    const int gwave = blockIdx.x * 8 + wave;

    const int half = lane >> 4;   // 0: lanes 0-15, 1: lanes 16-31
    const int l16  = lane & 15;

    // ---- A operands (components, loop-invariant): 4 M-tiles x 2 K-chunks.
    // 16x4 f32 A layout: lane holds M = l16 (component j = 16t + l16),
    // VGPR p holds K = 4c + 2*half + p.
    v2f bA[4][2];
    #pragma unroll
    for (int t = 0; t < 4; ++t) {
        const int col = t*16 + l16;
        #pragma unroll
        for (int c = 0; c < 2; ++c) {
            const int kb = c*4 + half*2;
            bA[t][c].x = Bt[col*8 + kb + 0];
            bA[t][c].y = Bt[col*8 + kb + 1];
        }
    }

    #pragma unroll
    for (int it = 0; it < ITERS; ++it) {
        const int base = (gwave * ITERS + it) * 16;
        if (base >= N_PTS) break;                    // wave-uniform

        // ---- B operand (points): 4x16 f32 layout, N = l16, VGPR p holds
        // K = 2*half + p (chunk 0) / 4 + 2*half + p (chunk 1).
        int row = base + l16;
        row = row < N_PTS ? row : (N_PTS - 1);       // clamp; EXEC stays all-1
        const v2f x = ((const v2f*)sample)[row];
        const float x0 = x.x, x1 = x.y;

        v2f p0, p1;
        p0.x = half ? (x0*x1) : (x0*x0);   // K = 0 / 2
        p0.y = half ? (x1*x1) : (x0*x1);   // K = 1 / 3
        p1.x = half ? 1.0f    : x0;        // K = 4 / 6
        p1.y = half ? 0.0f    : x1;        // K = 5 / 7

        // ---- 8 WMMAs: 4 component tiles, K=8 as two x4 chunks.
        v8f acc[4];
        #pragma unroll
        for (int t = 0; t < 4; ++t) {
            v8f c = {};
            c = __builtin_amdgcn_wmma_f32_16x16x4_f32(false, bA[t][0], false, p0,
                                                      (short)0, c, false, false);
            c = __builtin_amdgcn_wmma_f32_16x16x4_f32(false, bA[t][1], false, p1,
                                                      (short)0, c, false, false);
            acc[t] = c;
        }

        // ---- In-lane logsumexp over this lane's 32 components, then a single
        // cross-half exchange covers all 64.
        float m = acc[0][0];
        #pragma unroll
        for (int t = 0; t < 4; ++t)
            #pragma unroll
            for (int r = 0; r < 8; ++r)
                m = fmaxf(m, acc[t][r]);
        m = fmaxf(m, __shfl_xor(m, 16, 32));

        float s = 0.0f;
        #pragma unroll
        for (int t = 0; t < 4; ++t)
            #pragma unroll
            for (int r = 0; r < 8; ++r)
                s += __builtin_amdgcn_exp2f(acc[t][r] - m);
        s += __shfl_xor(s, 16, 32);

        const int prow = base + l16;
        if (half == 0 && prow < N_PTS)
            out[prow] = (m + log2f(s)) * LN2f;
    }
}

extern "C" void kernel_launch(void* const* d_in, const int* in_sizes, int n_in,
                              void* d_out, int out_size, void* d_ws, size_t ws_size,
                              hipStream_t stream) {
    const float* sample = (const float*)d_in[0];   // (1e6, 2)
    const float* mu     = (const float*)d_in[1];   // (64, 2)
    const float* A      = (const float*)d_in[2];   // (64, 2, 2)
    const float* w      = (const float*)d_in[3];   // (64, 1)
    float* out = (float*)d_out;                    // (1e6, 1)
    float* Bt  = (float*)d_ws;                     // 64*8 floats = 2 KB

    gmm_prep<<<1, 64, 0, stream>>>(mu, A, w, Bt);

    const int ptsPerBlock = 8 * ITERS * 16;                    // 512
    const int nBlocks = (N_PTS + ptsPerBlock - 1) / ptsPerBlock; // 1954
    gmm_main<<<nBlocks, 256, 0, stream>>>(sample, Bt, out);
}